// SingleGcnLayer_36515811951250
// MI455X (gfx1250) — compile-verified
//
#include <hip/hip_runtime.h>

// Problem constants (match reference)
#define B_ 32
#define N_ 1024
#define KNBR 16
#define D_ 256
#define L_ 256

typedef float v2f __attribute__((ext_vector_type(2)));
typedef float v4f __attribute__((ext_vector_type(4)));
typedef float v8f __attribute__((ext_vector_type(8)));

// Padded LDS row stride in floats. 260*4 = 1040 bytes:
//  - multiple of 16 -> b128 stores and b64 fragment loads stay aligned
//  - bank(row m) = (260*m) mod 64 = 4m mod 64 -> all 16 rows hit distinct banks
#define SROW 260

__global__ __launch_bounds__(512, 2)
void gcn_fused_wmma(const float* __restrict__ targets,
                    const float* __restrict__ neighbors,
                    const float* __restrict__ a,
                    const float* __restrict__ kern,
                    float* __restrict__ out)
{
    __shared__ alignas(16) float sS[16 * SROW]; // S = targets + sum_k neighbors
    __shared__ float sInvF[16];                 // 1 / degree factor per row

    const int tid = threadIdx.x;                // 0..511 (16 wave32 waves)
    const long long row0 = (long long)blockIdx.x * 16;

    // ---------------- Phase 1: streaming neighborhood reduction (b128) ----------------
    // 16 rows x 64 float4 = 1024 vec4 elements, 2 per thread; consecutive tid -> contiguous 16B
    {
        const v4f* tg4 = (const v4f*)targets;
        const v4f* nb4 = (const v4f*)neighbors;
        #pragma unroll
        for (int e = 0; e < 2; ++e) {
            const int f4 = tid + e * 512;       // 0..1023
            const int m  = f4 >> 6;             // row in tile (64 vec4 per row)
            const int d4 = f4 & 63;
            const long long r = row0 + m;       // global (b*N+n) row
            v4f acc = __builtin_nontemporal_load(&tg4[r * (D_ / 4) + d4]);
            const v4f* nb = &nb4[(r * KNBR) * (D_ / 4) + d4];
            #pragma unroll
            for (int k = 0; k < KNBR; ++k)
                acc += __builtin_nontemporal_load(&nb[k * (D_ / 4)]);
            *(v4f*)&sS[m * SROW + d4 * 4] = acc;
        }
    }
    if (tid < 16) {
        const long long r = row0 + tid;
        float f = 0.0f;
        #pragma unroll
        for (int j = 0; j < KNBR + 1; ++j)
            f += a[r * (KNBR + 1) + j];
        if (!(f > 0.5f)) f = 1.0f;
        sInvF[tid] = 1.0f / f;
    }
    __syncthreads();

    // ---------------- Phase 2: 16x16 tile per wave via V_WMMA_F32_16X16X4_F32 ----------------
    const int wave = tid >> 5;                  // 0..15 -> column tile
    const int lane = tid & 31;
    const int mn   = lane & 15;                 // A-row (=M) and B-col (=N) for this lane
    const int koff = (lane >> 4) << 1;          // lanes 16-31 hold K+2/K+3 per ISA A/B layouts
    const int col0 = wave * 16;

    v8f c = {};
    const float* Arow = &sS[mn * SROW];
    const float* Bcol = kern + col0 + mn;       // column (col0+mn), row-strided by L_

    #pragma unroll 8
    for (int kk = 0; kk < D_; kk += 4) {
        v2f Af, Bf;
        Af.x = Arow[kk + koff];
        Af.y = Arow[kk + koff + 1];
        Bf.x = Bcol[(kk + koff)     * L_];
        Bf.y = Bcol[(kk + koff + 1) * L_];
        // D = A(16x4) * B(4x16) + C : (neg_a, A, neg_b, B, c_mod, C, reuse_a, reuse_b)
        c = __builtin_amdgcn_wmma_f32_16x16x4_f32(false, Af, false, Bf,
                                                  (short)0, c, false, false);
    }

    // ---------------- Epilogue: degree-normalize + ReLU, coalesced stores ----------------
    const int halfBase = (lane >> 4) << 3;      // C/D: VGPR v holds row v + 8*(lane>>4)
    #pragma unroll
    for (int v = 0; v < 8; ++v) {
        const int m = v + halfBase;
        float val = c[v] * sInvF[m];
        val = val > 0.0f ? val : 0.0f;
        out[(row0 + m) * L_ + col0 + mn] = val;
    }
}

extern "C" void kernel_launch(void* const* d_in, const int* in_sizes, int n_in,
                              void* d_out, int out_size, void* d_ws, size_t ws_size,
                              hipStream_t stream) {
    const float* targets   = (const float*)d_in[0];  // [B,N,D]
    const float* neighbors = (const float*)d_in[1];  // [B,N,K,D]
    const float* a         = (const float*)d_in[2];  // [B,N,K+1]
    const float* kern      = (const float*)d_in[3];  // [D,L]
    float* out             = (float*)d_out;          // [B,N,L]

    const int nTiles = (B_ * N_) / 16;               // 2048 blocks
    gcn_fused_wmma<<<nTiles, 512, 0, stream>>>(targets, neighbors, a, kern, out);
}